// regressor_ffa_19052474925167
// MI455X (gfx1250) — compile-verified
//
#include <hip/hip_runtime.h>

// FFA pulsar-search pipeline for MI455X (gfx1250, wave32).
//  - stages 1..8 of each butterfly fused in LDS (256 rows x bins, row-per-thread)
//  - remaining global stages fused in PAIRS (roll composition) -> half the passes
//  - conv1->lrelu->conv2->lrelu->conv3 fused per 256-position tile; conv2 lowered
//    to V_WMMA_F32_16X16X4_F32 (M=16 ch, K=88=8ch*11taps, N=16 positions), K-loop
//    fully unrolled (22 wmma per tile)
//  - global argmax via order-preserving u64 key + global_atomic_max_u64

#define BATCH  8
#define NSAMP  (1 << 20)
#define NHALF  (1 << 19)
#define RTOT   491520
#define NPARTS 10
#define SEG    49152            // RTOT / NPARTS
#define SLOPE  0.01f
#define A1P    277              // 276 a1 columns + 1 pad

typedef __attribute__((ext_vector_type(2))) float v2f;
typedef __attribute__((ext_vector_type(8))) float v8f;
typedef unsigned long long u64;

#if defined(__has_builtin)
#if __has_builtin(__builtin_amdgcn_wmma_f32_16x16x4_f32)
#define USE_WMMA 1
#endif
#endif
#ifndef USE_WMMA
#define USE_WMMA 0
#endif

__device__ __forceinline__ float lrelu(float x) { return x > 0.0f ? x : SLOPE * x; }

// monotonic float -> u32 map (for max reduction incl. negatives)
__device__ __forceinline__ unsigned order_key(float f) {
  unsigned u = __float_as_uint(f);
  return (u & 0x80000000u) ? ~u : (u | 0x80000000u);
}

// ---------------------------------------------------------------- std (ddof=1)
__global__ __launch_bounds__(256) void std_kernel(const float* __restrict__ x,
                                                  float* __restrict__ stdv) {
  const int b = blockIdx.x;
  const int tid = threadIdx.x;
  const long base = (long)b * NSAMP;
  double s = 0.0, s2 = 0.0;
  for (int e = tid; e < NSAMP; e += 256) {
    double v = (double)x[base + e];
    s += v; s2 += v * v;
  }
  __shared__ double rs[256], rq[256];
  rs[tid] = s; rq[tid] = s2;
  __syncthreads();
  for (int o = 128; o > 0; o >>= 1) {
    if (tid < o) { rs[tid] += rs[tid + o]; rq[tid] += rq[tid + o]; }
    __syncthreads();
  }
  if (tid == 0) {
    double var = (rq[0] - rs[0] * rs[0] / (double)NSAMP) / (double)(NSAMP - 1);
    stdv[b] = (float)sqrt(var);
  }
}

// -------------------------------------------------- 2x linear downsample = pair avg
__global__ __launch_bounds__(256) void down_kernel(const float* __restrict__ x,
                                                   float* __restrict__ d) {
  const int i = blockIdx.x * 256 + threadIdx.x;    // < BATCH*NHALF
  const int b = i / NHALF;
  const int j = i - b * NHALF;
  const long base = (long)b * NSAMP + 2L * j;
  d[i] = 0.5f * (x[base] + x[base + 1]);
}

__global__ void init_best_kernel(u64* __restrict__ best) {
  if (threadIdx.x < BATCH) best[threadIdx.x] = 0ULL;
}

// ------------------------------------------- FFA stages 1..8 fused in LDS
template <int BINS>
__global__ __launch_bounds__(256) void ffa_first_kernel(
    const float* __restrict__ src, float* __restrict__ dst,
    int rowsValid, int target, int Nd) {
  __shared__ float bufA[256 * 25];
  __shared__ float bufB[256 * 25];
  const int tid = threadIdx.x;
  const int b = blockIdx.y;
  const int row0 = blockIdx.x << 8;
  const int nel = BINS << 8;
  const long sb = (long)b * Nd;
  // coalesced load (element-linear)
  for (int e = tid; e < nel; e += 256) {
    const int r = e / BINS, col = e - r * BINS;
    const int grow = row0 + r;
    bufA[r * 25 + col] = (grow < rowsValid) ? src[sb + (long)grow * BINS + col] : 0.0f;
  }
  __syncthreads();
  float* pin = bufA;
  float* pout = bufB;
  const int r = tid;                   // one row per thread
  for (int s = 1; s <= 8; ++s) {
    const int g = 1 << s, h = g >> 1;
    const int i = r & (g - 1);
    const int iA = (r - i) + (i >> 1);
    const int iB = iA + h;
    const int sh = ((i + 1) >> 1) % BINS;
    const float* pA = pin + iA * 25;
    const float* pB = pin + iB * 25;
    float* po = pout + r * 25;
#pragma unroll
    for (int col = 0; col < BINS; ++col) {
      int cc = col - sh; if (cc < 0) cc += BINS;
      po[col] = pA[col] + pB[cc];
    }
    __syncthreads();
    float* t = pin; pin = pout; pout = t;
  }
  const long ob = ((long)b * target + row0) * BINS;
  for (int e = tid; e < nel; e += 256) {
    const int rr = e / BINS, col = e - rr * BINS;
    dst[ob + e] = pin[rr * 25 + col];
  }
}

// --------------------------- FFA: two global stages (s, s+1) fused in one pass
template <int BINS>
__global__ __launch_bounds__(256) void ffa_stage2_kernel(
    const float* __restrict__ in, float* __restrict__ out, int s, int target) {
  const int e = blockIdx.x * 256 + threadIdx.x;
  const int per = target * BINS;
  const int b = e / per;
  const int rem = e - b * per;
  const int orow = rem / BINS;
  const int col = rem - orow * BINS;
  // stage s+1 decomposition of output row
  const int G = 1 << (s + 1), H = 1 << s;
  const int g = H, h = H >> 1;
  const int i1 = orow & (G - 1);
  const int iA = (orow - i1) + (i1 >> 1);
  const int iB = iA + H;
  const int sh1 = ((i1 + 1) >> 1) % BINS;
  // stage s decomposition of iA
  const int ia = iA & (g - 1);
  const int A0 = (iA - ia) + (ia >> 1);
  const int A1 = A0 + h;
  const int shA = ((ia + 1) >> 1) % BINS;
  // stage s decomposition of iB
  const int ib = iB & (g - 1);
  const int B0 = (iB - ib) + (ib >> 1);
  const int B1 = B0 + h;
  const int shB = ((ib + 1) >> 1) % BINS;
  // column rotations (compose mod BINS)
  int cA = col - shA;        if (cA < 0) cA += BINS;
  int cB = col - sh1;        if (cB < 0) cB += BINS;
  int cBB = cB - shB;        if (cBB < 0) cBB += BINS;
  const long rb = (long)b * per;
  // keep reference FP association: (a + rollA) + (b + rollB)
  const float left  = in[rb + (long)A0 * BINS + col] + in[rb + (long)A1 * BINS + cA];
  const float right = in[rb + (long)B0 * BINS + cB]  + in[rb + (long)B1 * BINS + cBB];
  out[rb + rem] = left + right;
}

// ------------------------------------------- FFA: single global stage (odd tail)
template <int BINS>
__global__ __launch_bounds__(256) void ffa_stage_kernel(
    const float* __restrict__ in, float* __restrict__ out, int s, int target) {
  const int e = blockIdx.x * 256 + threadIdx.x;
  const int per = target * BINS;
  const int b = e / per;
  const int rem = e - b * per;
  const int orow = rem / BINS;
  const int col = rem - orow * BINS;
  const int g = 1 << s, h = g >> 1;
  const int i = orow & (g - 1);
  const int iA = (orow - i) + (i >> 1);
  const int iB = iA + h;
  int c2 = col - (((i + 1) >> 1) % BINS);
  if (c2 < 0) c2 += BINS;
  const long rb = (long)b * per;
  out[rb + rem] = in[rb + (long)iA * BINS + col] + in[rb + (long)iB * BINS + c2];
}

// -------------------------------- per row: (max - median)/std/sqrt(rowsValid)
template <int BINS>
__global__ __launch_bounds__(256) void row_stats_kernel(
    const float* __restrict__ in, const float* __restrict__ stdv,
    float* __restrict__ snr, int roff, int target, int rowsValid) {
  const int idx = blockIdx.x * 256 + threadIdx.x;
  const int b = idx / target;
  const int row = idx - b * target;
  const float* p = in + ((long)b * target + row) * BINS;
  float v[BINS];
  float mx = -3.402823466e38f;
#pragma unroll
  for (int j = 0; j < BINS; ++j) { v[j] = p[j]; mx = fmaxf(mx, v[j]); }
  constexpr int M = (BINS - 1) / 2;   // sorted[(bins-1)//2]  (torch lower median)
  float med = v[0];
#pragma unroll
  for (int i = 0; i < BINS; ++i) {
    int less = 0, leq = 0;
#pragma unroll
    for (int j = 0; j < BINS; ++j) { less += (v[j] < v[i]); leq += (v[j] <= v[i]); }
    if (less <= M && M < leq) med = v[i];
  }
  snr[(long)b * RTOT + roff + row] = (mx - med) / stdv[b] / sqrtf((float)rowsValid);
}

// ------------------------------- per (b,part) standardization (ddof=1)
__global__ __launch_bounds__(256) void renorm_kernel(const float* __restrict__ snr,
                                                     float* __restrict__ tn) {
  const int part = blockIdx.x, b = blockIdx.y, tid = threadIdx.x;
  const long base = (long)b * RTOT + (long)part * SEG;
  double s = 0.0, s2 = 0.0;
  for (int e = tid; e < SEG; e += 256) {
    double v = (double)snr[base + e];
    s += v; s2 += v * v;
  }
  __shared__ double rs[256], rq[256];
  rs[tid] = s; rq[tid] = s2;
  __syncthreads();
  for (int o = 128; o > 0; o >>= 1) {
    if (tid < o) { rs[tid] += rs[tid + o]; rq[tid] += rq[tid + o]; }
    __syncthreads();
  }
  const float mean = (float)(rs[0] / (double)SEG);
  const float sd = (float)sqrt((rq[0] - rs[0] * rs[0] / (double)SEG) / (double)(SEG - 1));
  for (int e = tid; e < SEG; e += 256) tn[base + e] = (snr[base + e] - mean) / sd;
}

// ---- fused conv1(k11,p5) -> lrelu -> conv2(k11,d2,p10) [WMMA] -> lrelu -> conv3(1x1)
//      + block max/argmax + global atomic reduce
__global__ __launch_bounds__(256) void conv_argmax_kernel(
    const float* __restrict__ tn,
    const float* __restrict__ gW1, const float* __restrict__ gb1,
    const float* __restrict__ gW2, const float* __restrict__ gb2,
    const float* __restrict__ gW3, const float* __restrict__ gb3,
    u64* __restrict__ best) {
  __shared__ float s_tn[288];          // tnorm tile with halo [p0-15 .. p0+270]
  __shared__ float s_a1[8 * A1P];      // conv1+lrelu tile, q = pos-(p0-10) in [0,276)
  __shared__ float s_W1[88];
  __shared__ float s_b1[8];
  __shared__ float s_W2[16 * 88];      // [c2][c1*11+k]  (native layout of W2)
  __shared__ float s_b2[16];
  __shared__ float s_W3[16];
  __shared__ float s_b3;
  __shared__ u64 s_red[256];

  const int tid = threadIdx.x;
  const int b = blockIdx.y;
  const int p0 = blockIdx.x << 8;

  if (tid < 88) s_W1[tid] = gW1[tid];
  if (tid < 8) s_b1[tid] = gb1[tid];
  if (tid < 16) { s_b2[tid] = gb2[tid]; s_W3[tid] = gW3[tid]; }
  if (tid == 0) s_b3 = gb3[0];
  for (int e = tid; e < 16 * 88; e += 256) s_W2[e] = gW2[e];
  for (int e = tid; e < 288; e += 256) {
    const int pos = p0 - 15 + e;
    s_tn[e] = (pos >= 0 && pos < RTOT) ? tn[(long)b * RTOT + pos] : 0.0f;
  }
  __syncthreads();

  // conv1 + lrelu into LDS (zero outside [0,R) == conv2's zero padding of a1)
  for (int e = tid; e < 8 * 276; e += 256) {
    const int c1 = e / 276;
    const int q = e - c1 * 276;
    const int pos = p0 - 10 + q;
    float val = 0.0f;
    if (pos >= 0 && pos < RTOT) {
      float a = s_b1[c1];
#pragma unroll
      for (int j = 0; j < 11; ++j) a += s_W1[c1 * 11 + j] * s_tn[q + j];
      val = lrelu(a);
    }
    s_a1[c1 * A1P + q] = val;
  }
  __syncthreads();

  u64 mybest = 0ULL;
#if USE_WMMA
  // Each of the 8 waves handles 2 tiles of 16 positions: M=16 ch, K=88, N=16.
  const int lane = tid & 31;
  const int wv = tid >> 5;
  const int half = lane >> 4;          // 0: K pair {0,1}, 1: K pair {2,3}
  const int n = lane & 15;             // N (and A-matrix M) index
  for (int tt = 0; tt < 2; ++tt) {
    const int tbase = ((wv << 1) + tt) << 4;
    const int bcol = tbase + n;        // common B-fragment column offset
    v8f acc;
#pragma unroll
    for (int j = 0; j < 8; ++j) acc[j] = s_b2[j + (half << 3)];  // C VGPR j: M=j / j+8
#pragma unroll
    for (int kk = 0; kk < 22; ++kk) {
      const int k0 = (kk << 2) + (half << 1);   // A VGPR0 = K{0,2}, VGPR1 = K{1,3}
      const int k1 = k0 + 1;
      v2f af, bfrag;
      af.x = s_W2[n * 88 + k0];                                  // A[m][k0]
      af.y = s_W2[n * 88 + k1];                                  // A[m][k1]
      bfrag.x = s_a1[(k0 / 11) * A1P + bcol + 2 * (k0 % 11)];    // B[k0][n]
      bfrag.y = s_a1[(k1 / 11) * A1P + bcol + 2 * (k1 % 11)];    // B[k1][n]
      acc = __builtin_amdgcn_wmma_f32_16x16x4_f32(false, af, false, bfrag,
                                                  (short)0, acc, false, false);
    }
    // conv3: sum over 16 channels (split across the two half-waves)
    float sum = 0.0f;
#pragma unroll
    for (int j = 0; j < 8; ++j) sum += s_W3[j + (half << 3)] * lrelu(acc[j]);
    sum += __shfl_xor(sum, 16, 32);
    const float o3 = sum + s_b3;
    const unsigned p = (unsigned)(p0 + tbase + n);
    const u64 key = ((u64)order_key(o3) << 32) | (u64)(0xFFFFFFFFu - p);
    mybest = key > mybest ? key : mybest;
  }
#else
  {
    float sum = 0.0f;
    for (int c2 = 0; c2 < 16; ++c2) {
      float a = s_b2[c2];
#pragma unroll
      for (int k = 0; k < 88; ++k)
        a += s_W2[c2 * 88 + k] * s_a1[(k / 11) * A1P + tid + 2 * (k % 11)];
      sum += s_W3[c2] * lrelu(a);
    }
    const float o3 = sum + s_b3;
    const unsigned p = (unsigned)(p0 + tid);
    mybest = ((u64)order_key(o3) << 32) | (u64)(0xFFFFFFFFu - p);
  }
#endif
  s_red[tid] = mybest;
  __syncthreads();
  for (int o = 128; o > 0; o >>= 1) {
    if (tid < o) s_red[tid] = s_red[tid] > s_red[tid + o] ? s_red[tid] : s_red[tid + o];
    __syncthreads();
  }
  if (tid == 0) atomicMax(&best[b], s_red[0]);
}

// ---------------------------------------------------------------- head + period
__global__ void finalize_kernel(const u64* __restrict__ best,
                                const float* __restrict__ Wf,
                                const float* __restrict__ bfv,
                                float* __restrict__ out) {
  const int b = threadIdx.x;
  if (b >= BATCH) return;
  const u64 k = best[b];
  const unsigned u = (unsigned)(k >> 32);
  const unsigned p = 0xFFFFFFFFu - (unsigned)(k & 0xFFFFFFFFu);
  const unsigned bits = (u & 0x80000000u) ? (u ^ 0x80000000u) : ~u;
  const float v = __uint_as_float(bits);                 // pooled max
  out[b * 3 + 0] = Wf[0] * v + bfv[0];
  out[b * 3 + 1] = Wf[1] * v + bfv[1];
  const int off[10] = {0, 65536, 131072, 196608, 262144,
                       327680, 360448, 393216, 425984, 458752};
  const int tgt[10] = {65536, 65536, 65536, 65536, 65536,
                       32768, 32768, 32768, 32768, 32768};
  const int bn[10] = {20, 21, 22, 23, 24, 20, 21, 22, 23, 24};
  double per = 0.0;
  for (int c = 0; c < 10; ++c) {
    if ((int)p >= off[c] && (int)p < off[c] + tgt[c]) {
      const double ts = (c < 5) ? 0.000256 : 0.000512;
      const int i = (int)p - off[c];
      per = ts * (double)bn[c] + (double)i * ts / (double)(tgt[c] - 1);
    }
  }
  out[b * 3 + 2] = (float)per;
}

// ---------------------------------------------------------------- host
struct FfaCfg { int ds, bins, rows, target, nstage, roff; };
static const FfaCfg kCfg[10] = {
    {1, 20, 52428, 65536, 16, 0},
    {1, 21, 49932, 65536, 16, 65536},
    {1, 22, 47662, 65536, 16, 131072},
    {1, 23, 45590, 65536, 16, 196608},
    {1, 24, 43690, 65536, 16, 262144},
    {2, 20, 26214, 32768, 15, 327680},
    {2, 21, 24966, 32768, 15, 360448},
    {2, 22, 23831, 32768, 15, 393216},
    {2, 23, 22795, 32768, 15, 425984},
    {2, 24, 21845, 32768, 15, 458752},
};

template <int BINS>
static void run_ffa_config(const float* src, int Nd, const FfaCfg& cf,
                           float* wsA, float* wsB, const float* wsStd,
                           float* wsSnr, hipStream_t stream) {
  ffa_first_kernel<BINS><<<dim3(cf.target / 256, BATCH), 256, 0, stream>>>(
      src, wsA, cf.rows, cf.target, Nd);
  float* cur = wsA;
  float* nxt = wsB;
  const int total = BATCH * cf.target * BINS;
  int s = 9;
  for (; s + 1 <= cf.nstage; s += 2) {
    ffa_stage2_kernel<BINS><<<total / 256, 256, 0, stream>>>(cur, nxt, s, cf.target);
    float* t = cur; cur = nxt; nxt = t;
  }
  if (s == cf.nstage) {
    ffa_stage_kernel<BINS><<<total / 256, 256, 0, stream>>>(cur, nxt, s, cf.target);
    float* t = cur; cur = nxt; nxt = t;
  }
  row_stats_kernel<BINS><<<(BATCH * cf.target) / 256, 256, 0, stream>>>(
      cur, wsStd, wsSnr, cf.roff, cf.target, cf.rows);
}

extern "C" void kernel_launch(void* const* d_in, const int* in_sizes, int n_in,
                              void* d_out, int out_size, void* d_ws, size_t ws_size,
                              hipStream_t stream) {
  (void)in_sizes; (void)n_in; (void)out_size; (void)ws_size;
  const float* x  = (const float*)d_in[0];
  const float* W1 = (const float*)d_in[1];
  const float* b1 = (const float*)d_in[2];
  const float* W2 = (const float*)d_in[3];
  const float* b2 = (const float*)d_in[4];
  const float* W3 = (const float*)d_in[5];
  const float* b3 = (const float*)d_in[6];
  const float* Wf = (const float*)d_in[7];
  const float* bf = (const float*)d_in[8];
  float* out = (float*)d_out;

  char* ws = (char*)d_ws;
  float* wsDown = (float*)(ws + 0);           // 16 MiB  [8][524288]
  float* wsA    = (float*)(ws + 16777216);    // 48 MiB  ping  [8][target][bins]
  float* wsB    = (float*)(ws + 67108864);    // 48 MiB  pong
  float* wsSnr  = (float*)(ws + 117440512);   // 15 MiB  [8][RTOT]
  float* wsStd  = (float*)(ws + 133169152);   // 8 floats
  u64*   wsBest = (u64*)  (ws + 133169408);   // 8 u64
  float* wsTn   = wsA;                        // renorm output reuses ping buffer

  std_kernel<<<BATCH, 256, 0, stream>>>(x, wsStd);
  down_kernel<<<(BATCH * NHALF) / 256, 256, 0, stream>>>(x, wsDown);
  init_best_kernel<<<1, 32, 0, stream>>>(wsBest);

  for (int c = 0; c < 10; ++c) {
    const FfaCfg cf = kCfg[c];
    const float* src = (cf.ds == 1) ? x : wsDown;
    const int Nd = (cf.ds == 1) ? NSAMP : NHALF;
    switch (cf.bins) {
      case 20: run_ffa_config<20>(src, Nd, cf, wsA, wsB, wsStd, wsSnr, stream); break;
      case 21: run_ffa_config<21>(src, Nd, cf, wsA, wsB, wsStd, wsSnr, stream); break;
      case 22: run_ffa_config<22>(src, Nd, cf, wsA, wsB, wsStd, wsSnr, stream); break;
      case 23: run_ffa_config<23>(src, Nd, cf, wsA, wsB, wsStd, wsSnr, stream); break;
      case 24: run_ffa_config<24>(src, Nd, cf, wsA, wsB, wsStd, wsSnr, stream); break;
    }
  }

  renorm_kernel<<<dim3(NPARTS, BATCH), 256, 0, stream>>>(wsSnr, wsTn);
  conv_argmax_kernel<<<dim3(RTOT / 256, BATCH), 256, 0, stream>>>(
      wsTn, W1, b1, W2, b2, W3, b3, wsBest);
  finalize_kernel<<<1, 32, 0, stream>>>(wsBest, Wf, bf, out);
}